// MambaMomentum_22978075034395
// MI455X (gfx1250) — compile-verified
//
#include <hip/hip_runtime.h>
#include <math.h>

// ---------------------------------------------------------------------------
// Mamba-momentum forward for MI455X (gfx1250, wave32, WMMA).
// All GEMMs use native f32 WMMA (V_WMMA_F32_16X16X4_F32) with a software-
// pipelined, double-buffered K loop and 32-bit saddr-form addressing.
// ---------------------------------------------------------------------------

typedef __attribute__((ext_vector_type(2))) float v2f;
typedef __attribute__((ext_vector_type(8))) float v8f;

#define L_SEQ   2048
#define DMODEL  1024
#define EDIM    2048
#define NST     16
#define DTRANK  64
#define KCONV   4
#define BETA_C  0.6f

// One K-chunk = 16 consecutive k values = 4 WMMA steps.
template <int NT>
struct KChunk {
  v2f a[4];        // A fragments for 4 steps
  v2f b[4][NT];    // B fragments for 4 steps x NT n-tiles
};

template <int NT>
__device__ __forceinline__ void load_chunk(KChunk<NT>& c,
                                           const float* __restrict__ A,
                                           const float* __restrict__ W,
                                           int aoff, const int* boff, int k0) {
#pragma unroll
  for (int s = 0; s < 4; ++s) {
    c.a[s] = *(const v2f*)(A + aoff + k0 + 4 * s);
#pragma unroll
    for (int t = 0; t < NT; ++t)
      c.b[s][t] = *(const v2f*)(W + boff[t] + k0 + 4 * s);
  }
}

template <int NT>
__device__ __forceinline__ void compute_chunk(v8f* acc, const KChunk<NT>& c) {
#pragma unroll
  for (int s = 0; s < 4; ++s) {
#pragma unroll
    for (int t = 0; t < NT; ++t) {
      acc[t] = __builtin_amdgcn_wmma_f32_16x16x4_f32(
          false, c.a[s], false, c.b[s][t], (short)0, acc[t], false, false);
    }
  }
}

// ---------------------------------------------------------------------------
// f32 WMMA GEMM:  C[M,N] = A[M,K] @ W[N,K]^T   (i.e. X @ W.T)
// Each wave computes one 16 x (16*NT) output tile; block = 8 waves = 128 rows.
// K is compile-time, K % 32 == 0. EPI==1 -> softplus(acc + bias[col]).
//
// WMMA f32 16x16x4 VGPR layouts (CDNA5 ISA 7.12.2):
//   A (16x4):  lanes 0-15 -> M=lane, K={k0,k0+1};  lanes 16-31 -> K={k0+2,k0+3}
//   B (4x16):  lanes 0-15 -> N=lane, K={k0,k0+1};  lanes 16-31 -> K={k0+2,k0+3}
//   C/D:       VGPR j: lanes 0-15 -> M=j, N=lane;  lanes 16-31 -> M=j+8
// ---------------------------------------------------------------------------
template <int NT, int EPI, int K>
__global__ __launch_bounds__(256)
void wmma_gemm_f32(const float* __restrict__ A, const float* __restrict__ W,
                   float* __restrict__ C, const float* __restrict__ bias,
                   int M, int N, int lda, int ldw, int ldc) {
  static_assert(K % 32 == 0, "K must be a multiple of 32 for double buffering");
  const int lane = threadIdx.x & 31;
  const int wave = threadIdx.x >> 5;
  const int half = lane >> 4;   // which K-pair this lane holds
  const int lidx = lane & 15;   // M index (A) / N index (B,C)

  const int m0 = (blockIdx.y * 8 + wave) * 16;
  const int n0 = blockIdx.x * (16 * NT);
  if (m0 >= M || n0 >= N) return;

  // 32-bit element offsets from uniform bases -> saddr-form global loads.
  const int aoff = (m0 + lidx) * lda + 2 * half;
  int boff[NT];
#pragma unroll
  for (int t = 0; t < NT; ++t) boff[t] = (n0 + t * 16 + lidx) * ldw + 2 * half;

  const v8f zero = {};
  v8f acc[NT];
#pragma unroll
  for (int t = 0; t < NT; ++t) acc[t] = zero;

  KChunk<NT> c0, c1;

  // Software pipeline: load chunk i+1 while the 16 WMMAs of chunk i issue.
  load_chunk<NT>(c0, A, W, aoff, boff, 0);
  for (int k0 = 0; k0 < K - 32; k0 += 32) {
    load_chunk<NT>(c1, A, W, aoff, boff, k0 + 16);
    compute_chunk<NT>(acc, c0);
    load_chunk<NT>(c0, A, W, aoff, boff, k0 + 32);
    compute_chunk<NT>(acc, c1);
  }
  load_chunk<NT>(c1, A, W, aoff, boff, K - 16);
  compute_chunk<NT>(acc, c0);
  compute_chunk<NT>(acc, c1);

#pragma unroll
  for (int t = 0; t < NT; ++t) {
    const int col = n0 + t * 16 + lidx;
#pragma unroll
    for (int j = 0; j < 8; ++j) {
      const int row = m0 + half * 8 + j;
      float v = acc[t][j];
      if (EPI == 1) {                       // softplus(v + bias)
        v += bias[col];
        v = (v > 20.0f) ? v : log1pf(__expf(v));
      }
      C[(size_t)row * ldc + col] = v;
    }
  }
}

// ---------------------------------------------------------------------------
// Depthwise causal conv1d (K=4) + bias + SiLU.
// xz is (L, 2*EDIM); the conv input is its first EDIM columns.
// ---------------------------------------------------------------------------
__global__ __launch_bounds__(256)
void conv_silu_kernel(const float* __restrict__ xz, const float* __restrict__ cw,
                      const float* __restrict__ cb, float* __restrict__ xc) {
  const int idx = blockIdx.x * blockDim.x + threadIdx.x;  // t*EDIM + e
  if (idx >= L_SEQ * EDIM) return;
  const int e = idx & (EDIM - 1);
  const int t = idx >> 11;  // /EDIM
  float s = cb[e];
#pragma unroll
  for (int k = 0; k < KCONV; ++k) {
    const int ts = t - (KCONV - 1) + k;
    if (ts >= 0) s += xz[(size_t)ts * (2 * EDIM) + e] * cw[e * KCONV + k];
  }
  xc[idx] = s / (1.0f + __expf(-s));  // silu
}

// ---------------------------------------------------------------------------
// Fused momentum scan. Each wave32 = 2 channels x 16 states (lane = esub*16+n).
// Per step: dA = exp(delta*A), u = delta*B*xc; v = beta*v + u; h = dA*h + v;
// y[t,e] = sum_n h*C (16-lane shfl_xor reduction) + D*xc; yg = y * silu(z).
// Avoids materializing deltaA/u/h (3 x 268 MB in the reference).
// ---------------------------------------------------------------------------
__global__ __launch_bounds__(256)
void scan_fused_kernel(const float* __restrict__ delta,  // (L, EDIM)
                       const float* __restrict__ dBC,    // (L, 96)
                       const float* __restrict__ xc,     // (L, EDIM)
                       const float* __restrict__ xz,     // (L, 2*EDIM); z = cols [EDIM, 2*EDIM)
                       const float* __restrict__ A_log,  // (EDIM, NST)
                       const float* __restrict__ Dv,     // (EDIM,)
                       float* __restrict__ yg) {         // (L, EDIM)
  const int lane = threadIdx.x & 31;
  const int wave = threadIdx.x >> 5;
  const int n = lane & 15;
  const int esub = lane >> 4;
  const int e = blockIdx.x * 16 + wave * 2 + esub;

  const float Aen = -__expf(A_log[e * NST + n]);
  const float De = Dv[e];

  float h = 0.0f, v = 0.0f;
  for (int t = 0; t < L_SEQ; ++t) {
    const float dl = delta[(size_t)t * EDIM + e];
    const float xt = xc[(size_t)t * EDIM + e];
    const float Bt = dBC[t * 96 + DTRANK + n];
    const float Ct = dBC[t * 96 + DTRANK + NST + n];

    const float dA = __expf(dl * Aen);
    const float u = dl * Bt * xt;     // ALPHA == 1
    v = BETA_C * v + u;
    h = dA * h + v;

    float p = h * Ct;                 // reduce over 16 states
    p += __shfl_xor(p, 1, 32);
    p += __shfl_xor(p, 2, 32);
    p += __shfl_xor(p, 4, 32);
    p += __shfl_xor(p, 8, 32);

    if (n == 0) {
      const float y = p + De * xt;
      const float z = xz[(size_t)t * (2 * EDIM) + EDIM + e];
      yg[(size_t)t * EDIM + e] = y * (z / (1.0f + __expf(-z)));
    }
  }
}

// ---------------------------------------------------------------------------
// Launch: 6 stages on `stream`, graph-capture safe (no alloc/sync/memcpy).
// Inputs (setup_inputs order): x, W_in, conv_w, conv_b, W_x, W_dt, b_dt,
//                              A_log, D, W_out
// ---------------------------------------------------------------------------
extern "C" void kernel_launch(void* const* d_in, const int* in_sizes, int n_in,
                              void* d_out, int out_size, void* d_ws, size_t ws_size,
                              hipStream_t stream) {
  const float* x      = (const float*)d_in[0];
  const float* W_in   = (const float*)d_in[1];
  const float* conv_w = (const float*)d_in[2];
  const float* conv_b = (const float*)d_in[3];
  const float* W_x    = (const float*)d_in[4];
  const float* W_dt   = (const float*)d_in[5];
  const float* b_dt   = (const float*)d_in[6];
  const float* A_log  = (const float*)d_in[7];
  const float* Dv     = (const float*)d_in[8];
  const float* W_out  = (const float*)d_in[9];
  float* out = (float*)d_out;

  // Workspace layout (floats): total ~21.2M floats (~81 MB)
  float* ws    = (float*)d_ws;
  float* xz    = ws;                                  // L * 2*EDIM  = 8.39M
  float* xc    = xz    + (size_t)L_SEQ * 2 * EDIM;    // L * EDIM    = 4.19M
  float* dBC   = xc    + (size_t)L_SEQ * EDIM;        // L * 96      = 0.20M
  float* delta = dBC   + (size_t)L_SEQ * 96;          // L * EDIM    = 4.19M
  float* yg    = delta + (size_t)L_SEQ * EDIM;        // L * EDIM    = 4.19M

  const dim3 blk(256, 1, 1);

  // 1) xz = x @ W_in^T : M=2048, N=4096, K=1024  (17.2 GFLOP)
  wmma_gemm_f32<4, 0, DMODEL><<<dim3(4096 / 64, L_SEQ / 128), blk, 0, stream>>>(
      x, W_in, xz, nullptr, L_SEQ, 2 * EDIM, DMODEL, DMODEL, 2 * EDIM);

  // 2) xc = silu(causal_dwconv(xz[:, :EDIM]) + conv_b)
  conv_silu_kernel<<<(L_SEQ * EDIM) / 256, blk, 0, stream>>>(xz, conv_w, conv_b, xc);

  // 3) dBC = xc @ W_x^T : M=2048, N=96, K=2048
  wmma_gemm_f32<2, 0, EDIM><<<dim3(96 / 32, L_SEQ / 128), blk, 0, stream>>>(
      xc, W_x, dBC, nullptr, L_SEQ, DTRANK + 2 * NST, EDIM, EDIM, DTRANK + 2 * NST);

  // 4) delta = softplus(dBC[:, :64] @ W_dt^T + b_dt) : M=2048, N=2048, K=64
  wmma_gemm_f32<4, 1, DTRANK><<<dim3(EDIM / 64, L_SEQ / 128), blk, 0, stream>>>(
      dBC, W_dt, delta, b_dt, L_SEQ, EDIM, DTRANK + 2 * NST, DTRANK, EDIM);

  // 5) fused momentum scan + C-reduction + D skip + silu(z) gating
  scan_fused_kernel<<<EDIM / 16, blk, 0, stream>>>(delta, dBC, xc, xz, A_log, Dv, yg);

  // 6) out = yg @ W_out^T : M=2048, N=1024, K=2048  (8.6 GFLOP)
  wmma_gemm_f32<4, 0, EDIM><<<dim3(DMODEL / 64, L_SEQ / 128), blk, 0, stream>>>(
      yg, W_out, out, nullptr, L_SEQ, DMODEL, EDIM, EDIM, DMODEL);

  (void)in_sizes; (void)n_in; (void)out_size; (void)ws_size;
}